// GNNModel_21002390078175
// MI455X (gfx1250) — compile-verified
//
#include <hip/hip_runtime.h>

typedef __attribute__((ext_vector_type(2))) float v2f;
typedef __attribute__((ext_vector_type(8))) float v8f;

#define FIN 6
#define HID 16

// ---------------- zero workspace (degs + agg buffers) ----------------
__global__ __launch_bounds__(256) void zero_kernel(float* __restrict__ p, int n) {
  int i = blockIdx.x * blockDim.x + threadIdx.x;
  if (i < n) p[i] = 0.0f;
}

// ---------------- degree accumulation ----------------
__global__ __launch_bounds__(256) void degree_kernel(const int* __restrict__ src,
                                                     const int* __restrict__ dst,
                                                     float* deg_out, float* deg_in, int E) {
  int e = blockIdx.x * blockDim.x + threadIdx.x;
  if (e >= E) return;
  atomicAdd(deg_out + src[e], 1.0f);
  atomicAdd(deg_in + dst[e], 1.0f);
}

// ---------------- norms (in-place over degs) + hs1 = h * norm_out ----------------
__global__ __launch_bounds__(256) void norm_hs1_kernel(const float* __restrict__ h,
                                                       float* deg_out, float* deg_in,
                                                       float* __restrict__ hs1, int N) {
  int i = blockIdx.x * blockDim.x + threadIdx.x;
  if (i >= N) return;
  float dO = deg_out[i];
  float dI = deg_in[i];
  float no = dO > 0.0f ? rsqrtf(dO) : 0.0f;
  float ni = dI > 0.0f ? rsqrtf(dI) : 0.0f;
  deg_out[i] = no;   // becomes norm_out
  deg_in[i]  = ni;   // becomes norm_in
#pragma unroll
  for (int k = 0; k < FIN; ++k) hs1[i * FIN + k] = h[i * FIN + k] * no;
}

// ---------------- layer-1 edge scatter: agg1[dst] += hs1[src] (6 wide) ----------------
__global__ __launch_bounds__(256) void scatter1_kernel(const int* __restrict__ src,
                                                       const int* __restrict__ dst,
                                                       const float* __restrict__ hs1,
                                                       float* agg1, int E) {
  int e = blockIdx.x * blockDim.x + threadIdx.x;
  if (e >= E) return;
  int s = src[e], d = dst[e];
  const float2* hp = (const float2*)(hs1 + (size_t)s * FIN);   // 24B row, 8B aligned
  float2 a = hp[0], b = hp[1], c = hp[2];
  float* ap = agg1 + (size_t)d * FIN;
  atomicAdd(ap + 0, a.x); atomicAdd(ap + 1, a.y);
  atomicAdd(ap + 2, b.x); atomicAdd(ap + 3, b.y);
  atomicAdd(ap + 4, c.x); atomicAdd(ap + 5, c.y);
}

// ---------------- layer-1 combine: hs2 = relu(agg1*ni @ W1 + b1) * no ----------------
__global__ __launch_bounds__(256) void combine1_kernel(const float* __restrict__ agg1,
                                                       const float* __restrict__ norm_in,
                                                       const float* __restrict__ norm_out,
                                                       const float* __restrict__ W1,
                                                       const float* __restrict__ b1,
                                                       float* __restrict__ hs2, int N) {
  int i = blockIdx.x * blockDim.x + threadIdx.x;
  if (i >= N) return;
  float ni = norm_in[i], no = norm_out[i];
  float v[FIN];
#pragma unroll
  for (int k = 0; k < FIN; ++k) v[k] = agg1[(size_t)i * FIN + k] * ni;
#pragma unroll
  for (int j = 0; j < HID; ++j) {
    float acc = b1[j];
#pragma unroll
    for (int k = 0; k < FIN; ++k) acc = fmaf(v[k], W1[k * HID + j], acc);
    acc = fmaxf(acc, 0.0f);
    hs2[(size_t)i * HID + j] = acc * no;
  }
}

// ---------------- layer-2 edge scatter: agg2[dst] += hs2[src] (16 wide) ----------------
__global__ __launch_bounds__(256) void scatter2_kernel(const int* __restrict__ src,
                                                       const int* __restrict__ dst,
                                                       const float* __restrict__ hs2,
                                                       float* agg2, int E) {
  int e = blockIdx.x * blockDim.x + threadIdx.x;
  if (e >= E) return;
  int s = src[e], d = dst[e];
  const float4* hp = (const float4*)(hs2 + (size_t)s * HID);   // 64B row, 16B aligned
  float4 v0 = hp[0], v1 = hp[1], v2 = hp[2], v3 = hp[3];
  float* ap = agg2 + (size_t)d * HID;
  atomicAdd(ap +  0, v0.x); atomicAdd(ap +  1, v0.y);
  atomicAdd(ap +  2, v0.z); atomicAdd(ap +  3, v0.w);
  atomicAdd(ap +  4, v1.x); atomicAdd(ap +  5, v1.y);
  atomicAdd(ap +  6, v1.z); atomicAdd(ap +  7, v1.w);
  atomicAdd(ap +  8, v2.x); atomicAdd(ap +  9, v2.y);
  atomicAdd(ap + 10, v2.z); atomicAdd(ap + 11, v2.w);
  atomicAdd(ap + 12, v3.x); atomicAdd(ap + 13, v3.y);
  atomicAdd(ap + 14, v3.z); atomicAdd(ap + 15, v3.w);
}

// ---------------- layer-2 combine via WMMA: h2 = relu((agg2*ni) @ W2 + b2), in-place ----
// One wave32 per 16-row tile. K=16 as 4 steps of V_WMMA_F32_16X16X4_F32.
// ISA f32 layouts (7.12.2): A 16x4: lanes 0-15 = {K0,K1}, lanes 16-31 = {K2,K3};
// B 4x16 mirrored; C/D: vgpr r, lane<16 -> (m=r, n=lane), lane>=16 -> (m=r+8, n=lane-16).
__global__ __launch_bounds__(256) void combine2_wmma_kernel(float* agg2,
                                                            const float* __restrict__ norm_in,
                                                            const float* __restrict__ W2,
                                                            const float* __restrict__ b2,
                                                            int numTiles) {
  int wave = (int)((blockIdx.x * blockDim.x + threadIdx.x) >> 5);
  int lane = threadIdx.x & 31;
  if (wave >= numTiles) return;           // uniform per wave: EXEC all-1s inside
  int row0 = wave * 16;
  int half = lane >> 4;                   // 0: K pair {0,1}; 1: K pair {2,3}
  int lr   = lane & 15;

  int arow = row0 + lr;
  float ni = norm_in[arow];
  const float* Arow = agg2 + (size_t)arow * HID;

  v8f c = {};
#pragma unroll
  for (int kb = 0; kb < 16; kb += 4) {
    int k0 = kb + half * 2;
    v2f a, b;
    a.x = Arow[k0 + 0] * ni;
    a.y = Arow[k0 + 1] * ni;
    b.x = W2[(k0 + 0) * HID + lr];
    b.y = W2[(k0 + 1) * HID + lr];
    c = __builtin_amdgcn_wmma_f32_16x16x4_f32(false, a, false, b, (short)0, c,
                                              false, false);
  }

  float bias = b2[lr];
#pragma unroll
  for (int r = 0; r < 8; ++r) {
    int m = r + half * 8;
    float val = fmaxf(c[r] + bias, 0.0f);
    agg2[(size_t)(row0 + m) * HID + lr] = val;   // in-place: tile rows are wave-private
  }
}

// scalar tail (only launched if N % 16 != 0; N=100000 -> not launched)
__global__ __launch_bounds__(256) void combine2_tail_kernel(float* agg2,
                                                            const float* __restrict__ norm_in,
                                                            const float* __restrict__ W2,
                                                            const float* __restrict__ b2,
                                                            int rowStart, int N) {
  int i = rowStart + blockIdx.x * blockDim.x + threadIdx.x;
  if (i >= N) return;
  float ni = norm_in[i];
  float v[HID];
#pragma unroll
  for (int k = 0; k < HID; ++k) v[k] = agg2[(size_t)i * HID + k] * ni;
#pragma unroll
  for (int j = 0; j < HID; ++j) {
    float acc = b2[j];
#pragma unroll
    for (int k = 0; k < HID; ++k) acc = fmaf(v[k], W2[k * HID + j], acc);
    agg2[(size_t)i * HID + j] = fmaxf(acc, 0.0f);
  }
}

// ---------------- head: out[i] = h2[i,:] . Wl + bl ----------------
__global__ __launch_bounds__(256) void head_kernel(const float* __restrict__ h2,
                                                   const float* __restrict__ Wl,
                                                   const float* __restrict__ bl,
                                                   float* __restrict__ out, int N) {
  int i = blockIdx.x * blockDim.x + threadIdx.x;
  if (i >= N) return;
  const float4* hp = (const float4*)(h2 + (size_t)i * HID);
  const float4* wp = (const float4*)Wl;
  float acc = bl[0];
#pragma unroll
  for (int q = 0; q < 4; ++q) {
    float4 hv = hp[q], wv = wp[q];
    acc = fmaf(hv.x, wv.x, acc);
    acc = fmaf(hv.y, wv.y, acc);
    acc = fmaf(hv.z, wv.z, acc);
    acc = fmaf(hv.w, wv.w, acc);
  }
  out[i] = acc;
}

extern "C" void kernel_launch(void* const* d_in, const int* in_sizes, int n_in,
                              void* d_out, int out_size, void* d_ws, size_t ws_size,
                              hipStream_t stream) {
  const float* h  = (const float*)d_in[0];
  const float* W1 = (const float*)d_in[1];
  const float* b1 = (const float*)d_in[2];
  const float* W2 = (const float*)d_in[3];
  const float* b2 = (const float*)d_in[4];
  const float* Wl = (const float*)d_in[5];
  const float* bl = (const float*)d_in[6];
  const int* src  = (const int*)d_in[7];
  const int* dst  = (const int*)d_in[8];
  float* out = (float*)d_out;

  const int N = in_sizes[0] / FIN;
  const int E = in_sizes[7];

  // Workspace layout (floats). Zeroed region first (2N degs + 6N agg1 + 16N agg2).
  float* ws = (float*)d_ws;
  float* deg_out = ws;                      // N  -> becomes norm_out
  float* deg_in  = ws + (size_t)N;          // N  -> becomes norm_in
  float* agg1    = ws + 2 * (size_t)N;      // 6N
  float* agg2    = ws + 8 * (size_t)N;      // 16N (reused as h2 after WMMA combine)
  float* hs1     = ws + 24 * (size_t)N;     // 6N
  float* hs2     = ws + 30 * (size_t)N;     // 16N   total 46N floats (~18.4 MB)

  const int B = 256;
  int zn = 24 * N;
  zero_kernel<<<(zn + B - 1) / B, B, 0, stream>>>(ws, zn);

  degree_kernel<<<(E + B - 1) / B, B, 0, stream>>>(src, dst, deg_out, deg_in, E);
  norm_hs1_kernel<<<(N + B - 1) / B, B, 0, stream>>>(h, deg_out, deg_in, hs1, N);

  scatter1_kernel<<<(E + B - 1) / B, B, 0, stream>>>(src, dst, hs1, agg1, E);
  combine1_kernel<<<(N + B - 1) / B, B, 0, stream>>>(agg1, deg_in, deg_out, W1, b1, hs2, N);

  scatter2_kernel<<<(E + B - 1) / B, B, 0, stream>>>(src, dst, hs2, agg2, E);

  int numTiles = N / 16;
  if (numTiles > 0) {
    int threads = numTiles * 32;
    combine2_wmma_kernel<<<(threads + B - 1) / B, B, 0, stream>>>(agg2, deg_in, W2, b2, numTiles);
  }
  int rem = N - numTiles * 16;
  if (rem > 0) {
    combine2_tail_kernel<<<(rem + B - 1) / B, B, 0, stream>>>(agg2, deg_in, W2, b2,
                                                              numTiles * 16, N);
  }

  head_kernel<<<(N + B - 1) / B, B, 0, stream>>>(agg2, Wl, bl, out, N);
}